// PhysNetInteractionModule_60120952209948
// MI455X (gfx1250) — compile-verified
//
#include <hip/hip_runtime.h>
#include <math.h>

// PhysNet interaction module for MI455X (gfx1250), wave32, fp32 WMMA.
//
// Roofline: pair phase reads f_ij (128 MB, NT-streamed) once -> ~5.5us at
// 23.3 TB/s HBM; atom tables (~51 MB) stay resident in the 192 MB L2, so the
// xj gathers + f32 atomic segment-sum are L2-local. Node GEMMs (~7 GFLOP)
// run on V_WMMA_F32_16X16X4_F32 at full fp32 fidelity.
//
// Softplus uses the branch-free stable form max(x,0)+log(1+exp(-|x|)) with
// fast transcendental intrinsics (v_exp_f32/v_log_f32) instead of libm
// log1pf, which previously expanded to ~50 VALU ops + divergent branches.

typedef __attribute__((ext_vector_type(2))) float v2f;
typedef __attribute__((ext_vector_type(8))) float v8f;

__device__ __forceinline__ float sp(float x) {
    // stable, branch-free softplus
    return fmaxf(x, 0.0f) + __logf(1.0f + __expf(-fabsf(x)));
}

__device__ __forceinline__ v8f wmma_f32_k4(v2f a, v2f b, v8f c) {
    // D = A(16x4 f32) * B(4x16 f32) + C(16x16 f32)
    return __builtin_amdgcn_wmma_f32_16x16x4_f32(
        /*neg_a=*/false, a, /*neg_b=*/false, b,
        /*c_mod=*/(short)0, c, /*reuse_a=*/false, /*reuse_b=*/false);
}

#define LN2F 0.6931471805599453f
#define LDS_STRIDE 68  // 64 + 4 pad floats: conflict-free A-reads and C-writes

// ---------------------------------------------------------------------------
// Kernel A: per-atom  x = sp(emb) - ln2;  vacc = sp(x@Wi + bi);  xjb = sp(x@Wj + bj)
// One wave handles 16 atom rows: stage x in LDS, 2 GEMM passes of 4 N-tiles.
// ---------------------------------------------------------------------------
__global__ __launch_bounds__(256)
void physnet_node_pre(const float* __restrict__ emb,
                      const float* __restrict__ Wi, const float* __restrict__ bi,
                      const float* __restrict__ Wj, const float* __restrict__ bj,
                      float* __restrict__ vacc, float* __restrict__ xjb,
                      int nAtoms) {
    __shared__ float lds[8][16 * LDS_STRIDE];
    const int lane = threadIdx.x & 31;
    const int w    = threadIdx.x >> 5;
    const int lo   = lane & 15;
    const int hi   = lane >> 4;
    const int nb   = nAtoms >> 4;
    const int wstride = gridDim.x * 8;

    for (int blk = blockIdx.x * 8 + w; blk < nb; blk += wstride) {
        const int base = blk << 4;
        // stage x = softplus(emb) - ln2 into LDS tile [16 x 64]
        for (int i = lane; i < 16 * 64; i += 32) {
            int r = i >> 6, c = i & 63;
            lds[w][r * LDS_STRIDE + c] = sp(emb[(size_t)(base + r) * 64 + c]) - LN2F;
        }
        for (int pass = 0; pass < 2; ++pass) {
            const float* W  = pass ? Wj  : Wi;
            const float* bb = pass ? bj  : bi;
            float*       op = pass ? xjb : vacc;
            v8f acc[4] = {};
            #pragma unroll
            for (int s = 0; s < 4; ++s) {
                const int k0 = 4 * s + 2 * hi;
                v2f a = *(const v2f*)&lds[w][lo * LDS_STRIDE + k0];
                #pragma unroll
                for (int t = 0; t < 4; ++t) {
                    const int n = 16 * t + lo;
                    v2f b;
                    b.x = W[(size_t)k0 * 64 + n];
                    b.y = W[(size_t)(k0 + 1) * 64 + n];
                    acc[t] = wmma_f32_k4(a, b, acc[t]);
                }
            }
            #pragma unroll
            for (int t = 0; t < 4; ++t) {
                const int n = 16 * t + lo;
                const float bias = bb[n];
                #pragma unroll
                for (int j = 0; j < 8; ++j) {
                    const int m = j + 8 * hi;
                    op[(size_t)(base + m) * 64 + n] = sp(acc[t][j] + bias);
                }
            }
        }
    }
}

// ---------------------------------------------------------------------------
// Kernel B: pair phase.  f_prime = f @ G^T via WMMA (16 pairs = M dim),
// msg = xj[idx_j] * f_prime, atomic segment-sum into vacc[idx_i] (L2-resident).
// f_ij and pair_indices are single-use streams -> non-temporal loads so they
// don't evict the L2-resident atom tables.
// ---------------------------------------------------------------------------
__global__ __launch_bounds__(256)
void physnet_pairs(const int* __restrict__ pidx,   // [2, P]: row0=idx_i, row1=idx_j
                   const float* __restrict__ f,    // [P, 16]
                   const float* __restrict__ G,    // [64, 16]
                   const float* __restrict__ xjb,  // [N, 64]
                   float* __restrict__ vacc,       // [N, 64] accumulator (init = xi)
                   int nPairs) {
    const int lane = threadIdx.x & 31;
    const int lo   = lane & 15;
    const int hi   = lane >> 4;
    const int pb   = nPairs >> 4;
    const int wstride = gridDim.x * 8;

    for (int blk = blockIdx.x * 8 + (int)(threadIdx.x >> 5); blk < pb; blk += wstride) {
        const int p0 = blk << 4;
        v8f acc[4] = {};
        #pragma unroll
        for (int s = 0; s < 4; ++s) {
            const int k0 = 4 * s + 2 * hi;
            v2f a = __builtin_nontemporal_load(
                        (const v2f*)&f[(size_t)(p0 + lo) * 16 + k0]);  // f row, NT stream
            #pragma unroll
            for (int t = 0; t < 4; ++t) {
                const int c = 16 * t + lo;
                v2f b = *(const v2f*)&G[(size_t)c * 16 + k0];          // G row (contig)
                acc[t] = wmma_f32_k4(a, b, acc[t]);
            }
        }
        // per-lane row indices (rows m = 8*hi .. 8*hi+7 of this 16-pair block)
        int ii[8], ij[8];
        #pragma unroll
        for (int j = 0; j < 8; ++j) {
            const int m = j + 8 * hi;
            ii[j] = __builtin_nontemporal_load(&pidx[p0 + m]);
            ij[j] = __builtin_nontemporal_load(&pidx[nPairs + p0 + m]);
        }
        #pragma unroll
        for (int t = 0; t < 4; ++t) {
            const int n = 16 * t + lo;
            #pragma unroll
            for (int j = 0; j < 8; ++j) {
                const float msg = acc[t][j] * xjb[(size_t)ij[j] * 64 + n];
                atomicAdd(&vacc[(size_t)ii[j] * 64 + n], msg);
            }
        }
    }
}

// ---------------------------------------------------------------------------
// Kernel C: residual stack + output head on v = xi + summed.
//   for l in 0..2:  h = sp(v@W1+b1);  v += h@W2 + b2
//   out = sp(v) @ Wv + bv
// Stage boundaries round-trip through padded LDS tiles (C-layout -> A-layout).
// 128 threads = 4 waves per WG; 2 tiles/wave -> 34,816 B LDS.
// ---------------------------------------------------------------------------
__global__ __launch_bounds__(128)
void physnet_node_post(const float* __restrict__ vacc,
                       const float* __restrict__ rW1, const float* __restrict__ rb1,
                       const float* __restrict__ rW2, const float* __restrict__ rb2,
                       const float* __restrict__ Wv,  const float* __restrict__ bv,
                       float* __restrict__ out, int nAtoms) {
    __shared__ float ldsV[4][16 * LDS_STRIDE];
    __shared__ float ldsH[4][16 * LDS_STRIDE];
    const int lane = threadIdx.x & 31;
    const int w    = threadIdx.x >> 5;
    const int lo   = lane & 15;
    const int hi   = lane >> 4;
    const int nb   = nAtoms >> 4;
    const int wstride = gridDim.x * 4;

    for (int blk = blockIdx.x * 4 + w; blk < nb; blk += wstride) {
        const int base = blk << 4;
        for (int i = lane; i < 16 * 64; i += 32) {
            int r = i >> 6, c = i & 63;
            ldsV[w][r * LDS_STRIDE + c] = vacc[(size_t)(base + r) * 64 + c];
        }
        for (int l = 0; l < 3; ++l) {
            const float* W1 = rW1 + (size_t)l * 64 * 64;
            const float* b1 = rb1 + (size_t)l * 64;
            const float* W2 = rW2 + (size_t)l * 64 * 64;
            const float* b2 = rb2 + (size_t)l * 64;
            // h = sp(v @ W1 + b1)
            v8f acc[4] = {};
            #pragma unroll
            for (int s = 0; s < 4; ++s) {
                const int k0 = 4 * s + 2 * hi;
                v2f a = *(const v2f*)&ldsV[w][lo * LDS_STRIDE + k0];
                #pragma unroll
                for (int t = 0; t < 4; ++t) {
                    const int n = 16 * t + lo;
                    v2f b;
                    b.x = W1[(size_t)k0 * 64 + n];
                    b.y = W1[(size_t)(k0 + 1) * 64 + n];
                    acc[t] = wmma_f32_k4(a, b, acc[t]);
                }
            }
            #pragma unroll
            for (int t = 0; t < 4; ++t) {
                const int n = 16 * t + lo;
                const float bias = b1[n];
                #pragma unroll
                for (int j = 0; j < 8; ++j) {
                    const int m = j + 8 * hi;
                    ldsH[w][m * LDS_STRIDE + n] = sp(acc[t][j] + bias);
                }
            }
            // v += h @ W2 + b2
            v8f acd[4] = {};
            #pragma unroll
            for (int s = 0; s < 4; ++s) {
                const int k0 = 4 * s + 2 * hi;
                v2f a = *(const v2f*)&ldsH[w][lo * LDS_STRIDE + k0];
                #pragma unroll
                for (int t = 0; t < 4; ++t) {
                    const int n = 16 * t + lo;
                    v2f b;
                    b.x = W2[(size_t)k0 * 64 + n];
                    b.y = W2[(size_t)(k0 + 1) * 64 + n];
                    acd[t] = wmma_f32_k4(a, b, acd[t]);
                }
            }
            #pragma unroll
            for (int t = 0; t < 4; ++t) {
                const int n = 16 * t + lo;
                const float bias = b2[n];
                #pragma unroll
                for (int j = 0; j < 8; ++j) {
                    const int m = j + 8 * hi;
                    ldsV[w][m * LDS_STRIDE + n] += acd[t][j] + bias;
                }
            }
        }
        // out = sp(v) @ Wv + bv
        v8f acc[4] = {};
        #pragma unroll
        for (int s = 0; s < 4; ++s) {
            const int k0 = 4 * s + 2 * hi;
            v2f a = *(const v2f*)&ldsV[w][lo * LDS_STRIDE + k0];
            a.x = sp(a.x);
            a.y = sp(a.y);
            #pragma unroll
            for (int t = 0; t < 4; ++t) {
                const int n = 16 * t + lo;
                v2f b;
                b.x = Wv[(size_t)k0 * 64 + n];
                b.y = Wv[(size_t)(k0 + 1) * 64 + n];
                acc[t] = wmma_f32_k4(a, b, acc[t]);
            }
        }
        #pragma unroll
        for (int t = 0; t < 4; ++t) {
            const int n = 16 * t + lo;
            const float bias = bv[n];
            #pragma unroll
            for (int j = 0; j < 8; ++j) {
                const int m = j + 8 * hi;
                out[(size_t)(base + m) * 64 + n] = acc[t][j] + bias;
            }
        }
    }
}

// ---------------------------------------------------------------------------
extern "C" void kernel_launch(void* const* d_in, const int* in_sizes, int n_in,
                              void* d_out, int out_size, void* d_ws, size_t ws_size,
                              hipStream_t stream) {
    const int*   pidx = (const int*)d_in[0];     // [2, P]
    const float* f    = (const float*)d_in[1];   // [P, 1, 16]
    // d_in[2] = d_ij : unused by the reference
    const float* emb  = (const float*)d_in[3];   // [N, 64]
    const float* G    = (const float*)d_in[4];   // [64, 16]
    const float* Wi   = (const float*)d_in[5];
    const float* bi   = (const float*)d_in[6];
    const float* Wj   = (const float*)d_in[7];
    const float* bj   = (const float*)d_in[8];
    const float* rW1  = (const float*)d_in[9];   // [3, 64, 64]
    const float* rb1  = (const float*)d_in[10];  // [3, 64]
    const float* rW2  = (const float*)d_in[11];
    const float* rb2  = (const float*)d_in[12];
    const float* Wv   = (const float*)d_in[13];
    const float* bv   = (const float*)d_in[14];

    const int nPairs = in_sizes[0] / 2;
    const int nAtoms = in_sizes[3] / 64;

    // workspace: vacc [N*64] (xi + atomic message sum), xjb [N*64]
    float* vacc = (float*)d_ws;
    float* xjb  = vacc + (size_t)nAtoms * 64;

    const int nb = nAtoms / 16;  // 16-atom blocks per wave

    physnet_node_pre<<<(nb + 7) / 8, 256, 0, stream>>>(
        emb, Wi, bi, Wj, bj, vacc, xjb, nAtoms);

    physnet_pairs<<<1024, 256, 0, stream>>>(
        pidx, f, G, xjb, vacc, nPairs);

    physnet_node_post<<<(nb + 3) / 4, 128, 0, stream>>>(
        vacc, rW1, rb1, rW2, rb2, Wv, bv, (float*)d_out, nAtoms);
}